// GraphClassificationModelCrossEntropy_66288525247278
// MI455X (gfx1250) — compile-verified
//
#include <hip/hip_runtime.h>
#include <hip/hip_bf16.h>

typedef float v2f __attribute__((ext_vector_type(2)));
typedef float v8f __attribute__((ext_vector_type(8)));

#define IN_DIM  256
#define OUT_DIM 32
#define WSTRIDE 260  // padded K-stride for transposed W in LDS:
                     // even (8B-aligned b64 loads) and 260%64==4 -> lanes 0-15
                     // and 16-31 hit disjoint bank residue classes.

// ---------------------------------------------------------------------------
// 1) Degree counting: one thread per edge, two f32 atomics.
// ---------------------------------------------------------------------------
__global__ __launch_bounds__(256) void degree_kernel(
    const int* __restrict__ src, const int* __restrict__ dst,
    float* __restrict__ deg_out, float* __restrict__ deg_in, int n_edges) {
  int e = blockIdx.x * blockDim.x + threadIdx.x;
  if (e >= n_edges) return;
  atomicAdd(&deg_out[src[e]], 1.0f);
  atomicAdd(&deg_in[dst[e]], 1.0f);
}

// ---------------------------------------------------------------------------
// 2) norms: deg -> clip(deg,1)^-0.5 in place.
// ---------------------------------------------------------------------------
__global__ __launch_bounds__(256) void norm_kernel(
    float* __restrict__ deg_out, float* __restrict__ deg_in, int n_nodes) {
  int i = blockIdx.x * blockDim.x + threadIdx.x;
  if (i >= n_nodes) return;
  deg_out[i] = rsqrtf(fmaxf(deg_out[i], 1.0f));
  deg_in[i]  = rsqrtf(fmaxf(deg_in[i], 1.0f));
}

// ---------------------------------------------------------------------------
// 3) h = (x * norm_src[:,None]) @ W   via V_WMMA_F32_16X16X4_F32.
//    Block = 256 threads = 8 wave32; each wave computes rows [base,base+16)
//    x OUT_DIM=32 as two 16x16 accumulators, K=256 in steps of 4.
//    W is staged TRANSPOSED in LDS ([n][k], padded stride) so each lane's
//    B fragment is a single contiguous ds_load_b64 (no repack movs).
//    The per-row norm is a scalar => folded into the final D store.
// ---------------------------------------------------------------------------
__global__ __launch_bounds__(256) void gemm_wmma_kernel(
    const float* __restrict__ x, const float* __restrict__ W,
    const float* __restrict__ norm_src, float* __restrict__ h, int n_nodes) {
  __shared__ float WldsT[OUT_DIM * WSTRIDE];  // ~33 KB, [n][k] padded

  // Cooperative transpose: W is [k][n] row-major in global.
  for (int idx = threadIdx.x; idx < IN_DIM * OUT_DIM; idx += 256) {
    int k = idx >> 5;   // coalesced global read
    int n = idx & 31;
    WldsT[n * WSTRIDE + k] = W[idx];
  }
  __syncthreads();

  const int wave     = threadIdx.x >> 5;   // wave32
  const int lane     = threadIdx.x & 31;
  const int row_base = (blockIdx.x * 8 + wave) * 16;
  if (row_base >= n_nodes) return;

  const int lrow  = lane & 15;        // A row / B column handled by this lane
  const int khalf = (lane >> 4) * 2;  // lanes 0-15: K=k,k+1 ; lanes 16-31: K=k+2,k+3

  // Clamp out-of-range rows to a safe address; their D rows are discarded.
  int grow = row_base + lrow;
  int srow = (grow < n_nodes) ? grow : (n_nodes - 1);
  const float* xrow = x + (size_t)srow * IN_DIM;

  const float* wcol0 = &WldsT[lrow * WSTRIDE + khalf];         // B tile n=0..15
  const float* wcol1 = &WldsT[(16 + lrow) * WSTRIDE + khalf];  // B tile n=16..31

  v8f acc0 = {0.f, 0.f, 0.f, 0.f, 0.f, 0.f, 0.f, 0.f};
  v8f acc1 = {0.f, 0.f, 0.f, 0.f, 0.f, 0.f, 0.f, 0.f};

#pragma unroll 8
  for (int k = 0; k < IN_DIM; k += 4) {
    // A fragment (16x4 f32): b64 load of x[row, k+khalf .. k+khalf+1]
    float2 av = *(const float2*)(xrow + k + khalf);
    v2f a; a.x = av.x; a.y = av.y;

    // B fragments (4x16 f32): contiguous 8B-aligned ds_load_b64 each.
    v2f b0 = *(const v2f*)(wcol0 + k);
    v2f b1 = *(const v2f*)(wcol1 + k);

    // 8 args: (neg_a, A, neg_b, B, c_mod, C, reuse_a, reuse_b)
    acc0 = __builtin_amdgcn_wmma_f32_16x16x4_f32(false, a, false, b0,
                                                 (short)0, acc0, false, false);
    acc1 = __builtin_amdgcn_wmma_f32_16x16x4_f32(false, a, false, b1,
                                                 (short)0, acc1, false, false);
  }

  // D layout (16x16 f32): VGPR v -> M = v (lanes 0-15) or M = 8+v (lanes 16-31),
  // N = lane % 16. Apply per-row norm at the store (commutes with the GEMM).
  const int ncol  = lane & 15;
  const int mbase = (lane >> 4) * 8;
#pragma unroll
  for (int v = 0; v < 8; v++) {
    int r = row_base + mbase + v;
    if (r < n_nodes) {
      float nm = norm_src[r];
      h[(size_t)r * OUT_DIM + ncol]      = acc0[v] * nm;
      h[(size_t)r * OUT_DIM + 16 + ncol] = acc1[v] * nm;
    }
  }
}

// ---------------------------------------------------------------------------
// 4) Scatter: 8 lanes per edge; each lane moves one float4 of the 32-float
//    message with 4 global f32 atomics (L2-resident: h/out fit in 192MB L2).
// ---------------------------------------------------------------------------
__global__ __launch_bounds__(256) void scatter_kernel(
    const float* __restrict__ h, const int* __restrict__ src,
    const int* __restrict__ dst, float* __restrict__ out, int n_edges) {
  long long gid = (long long)blockIdx.x * blockDim.x + threadIdx.x;
  int e = (int)(gid >> 3);
  int q = (int)(gid & 7);
  if (e >= n_edges) return;
  int s = src[e];
  int d = dst[e];
  float4 v = *(const float4*)(h + (size_t)s * OUT_DIM + q * 4);
  float* o = out + (size_t)d * OUT_DIM + q * 4;
  atomicAdd(o + 0, v.x);
  atomicAdd(o + 1, v.y);
  atomicAdd(o + 2, v.z);
  atomicAdd(o + 3, v.w);
}

// ---------------------------------------------------------------------------
// 5) Finalize: out = out * norm_dst[:,None] + b
// ---------------------------------------------------------------------------
__global__ __launch_bounds__(256) void finalize_kernel(
    float* __restrict__ out, const float* __restrict__ norm_dst,
    const float* __restrict__ b, int n_nodes) {
  int i = blockIdx.x * blockDim.x + threadIdx.x;
  if (i >= n_nodes * OUT_DIM) return;
  int node = i >> 5;
  int j    = i & 31;
  out[i] = out[i] * norm_dst[node] + b[j];
}

extern "C" void kernel_launch(void* const* d_in, const int* in_sizes, int n_in,
                              void* d_out, int out_size, void* d_ws, size_t ws_size,
                              hipStream_t stream) {
  const float* x   = (const float*)d_in[0];
  const int*   src = (const int*)d_in[1];
  const int*   dst = (const int*)d_in[2];
  const float* W   = (const float*)d_in[3];
  const float* b   = (const float*)d_in[4];
  float*       out = (float*)d_out;

  const int N = in_sizes[0] / IN_DIM;  // 100000
  const int E = in_sizes[1];           // 1600000

  // Workspace layout: [deg_out/norm_src: N][deg_in/norm_dst: N][h: N*32]
  float* deg_out = (float*)d_ws;
  float* deg_in  = deg_out + N;
  float* h       = deg_in + N;

  hipMemsetAsync(d_ws, 0, (size_t)2 * N * sizeof(float), stream);
  hipMemsetAsync(d_out, 0, (size_t)out_size * sizeof(float), stream);

  degree_kernel<<<(E + 255) / 256, 256, 0, stream>>>(src, dst, deg_out, deg_in, E);
  norm_kernel<<<(N + 255) / 256, 256, 0, stream>>>(deg_out, deg_in, N);

  // 128 rows per block (8 waves x 16 rows)
  gemm_wmma_kernel<<<(N + 127) / 128, 256, 0, stream>>>(x, W, deg_out, h, N);

  long long scatter_threads = (long long)E * 8;
  scatter_kernel<<<(int)((scatter_threads + 255) / 256), 256, 0, stream>>>(h, src, dst, out, E);

  finalize_kernel<<<(N * OUT_DIM + 255) / 256, 256, 0, stream>>>(out, deg_in, b, N);
}